// BaselineDNN_42588895707994
// MI455X (gfx1250) — compile-verified
//
#include <hip/hip_runtime.h>
#include <hip/hip_bf16.h>

// Problem constants (DIM/OUT baked into the WMMA tiling; B/L derived at launch).
#define DIM 50        // embedding dim (K), padded to 64 in workspace
#define KPAD 64
#define OUT 20        // output logits (N), covered by two 16-wide N tiles
#define NPAD 32       // padded N for branchless B-fragment loads

typedef __attribute__((ext_vector_type(2))) float v2f;
typedef __attribute__((ext_vector_type(8))) float v8f;

// ---------------------------------------------------------------------------
// Kernel 0: zero-padded weight staging. Wpad[NPAD][KPAD]:
//   Wpad[n][k] = (n < OUT && k < DIM) ? W[n][k] : 0
// Makes every B-fragment load in the WMMA loop unconditional (padding rows/
// cols contribute exact zeros to the f32 accumulators).
// ---------------------------------------------------------------------------
__global__ __launch_bounds__(256)
void pad_w_kernel(const float* __restrict__ W, float* __restrict__ Wpad) {
    for (int i = threadIdx.x; i < NPAD * KPAD; i += 256) {
        const int n = i >> 6;       // 0..31
        const int k = i & (KPAD-1); // 0..63
        Wpad[i] = (n < OUT && k < DIM) ? W[n * DIM + k] : 0.0f;
    }
}

// ---------------------------------------------------------------------------
// Kernel 1: masked-mean embedding pooling + ReLU -> rep[B][KPAD] (zero padded)
// Block = (64, 4): x-dim covers embedding dims (50 active), y-dim strides tokens.
// ---------------------------------------------------------------------------
__global__ __launch_bounds__(256)
void pool_kernel(const int* __restrict__ x,
                 const int* __restrict__ lengths,
                 const float* __restrict__ emb,
                 float* __restrict__ rep,
                 int L) {
    const int row = blockIdx.x;
    const int d   = threadIdx.x;   // 0..63 (only 0..49 are real dims)
    const int ty  = threadIdx.y;   // 0..3 token stride
    const int len = lengths[row];  // >= 1 per reference

    const int* xrow = x + (size_t)row * (size_t)L;

    float acc = 0.0f;
    if (d < DIM) {
        for (int j = ty; j < len; j += 4) {
            const int tok = xrow[j];                         // broadcast within wave
            acc += emb[(size_t)tok * DIM + d];               // coalesced 200B/row
        }
    }

    __shared__ float smem[4][64];
    smem[ty][d] = acc;
    __syncthreads();

    if (ty == 0) {
        float s = smem[0][d] + smem[1][d] + smem[2][d] + smem[3][d];
        s = s / (float)len;
        s = fmaxf(s, 0.0f);                                  // ReLU
        rep[(size_t)row * KPAD + d] = (d < DIM) ? s : 0.0f;  // zero K-pad
    }
}

// ---------------------------------------------------------------------------
// Kernel 2: logits = rep @ W^T + b using V_WMMA_F32_16X16X4_F32.
// One wave per 16-row M tile (EXEC all-ones through the WMMA loop). 13 K-steps
// of 4 cover K=50 (zero pad to 52); two N tiles cover OUT=20 (zero pad to 32).
// All fragment loads are unconditional 8B-aligned b64s from padded buffers.
//
// Fragment layouts per CDNA5 ISA 7.12.2 (f32, 16x16x4):
//   A (16x4): lanes 0-15 -> M=lane, {v0,v1}={K0,K1}; lanes 16-31 -> {K2,K3}
//   B (4x16): mirrored over N: lanes 0-15 -> N=lane, {v0,v1}={K0,K1}; hi lanes K2,K3
//   C/D (16x16): VGPR r, lane l -> M = r + (l>=16 ? 8:0), N = l&15
// ---------------------------------------------------------------------------
__global__ __launch_bounds__(32)
void head_gemm_kernel(const float* __restrict__ rep,
                      const float* __restrict__ Wpad,  // [NPAD][KPAD], zero padded
                      const float* __restrict__ bias,  // [OUT]
                      float* __restrict__ out) {       // [B][OUT]
    const int lane  = threadIdx.x;         // 0..31
    const int row0  = blockIdx.x * 16;     // M tile origin
    const int mN    = lane & 15;           // M for A-frag, N for B-frag
    const int khalf = (lane >> 4) * 2;     // 0 for lanes 0-15, 2 for 16-31

    const float* arow  = rep  + (size_t)(row0 + mN) * KPAD + khalf;
    const float* brow0 = Wpad + (size_t)mN        * KPAD + khalf;  // N tile 0
    const float* brow1 = Wpad + (size_t)(16 + mN) * KPAD + khalf;  // N tile 1

    v8f c0 = {};   // N = 0..15
    v8f c1 = {};   // N = 16..31 (only 16..19 written back)

#pragma unroll
    for (int kk = 0; kk < 13; ++kk) {
        const int kb = kk * 4;                      // + khalf folded into bases
        v2f a  = *(const v2f*)(arow  + kb);
        v2f b0 = *(const v2f*)(brow0 + kb);
        v2f b1 = *(const v2f*)(brow1 + kb);

        c0 = __builtin_amdgcn_wmma_f32_16x16x4_f32(
                 false, a, false, b0, (short)0, c0, false, false);
        c1 = __builtin_amdgcn_wmma_f32_16x16x4_f32(
                 false, a, false, b1, (short)0, c1, false, false);
    }

    // Writeback: lane l covers column (l&15) [+16 for c1], rows mbase..mbase+7.
    const int n0 = mN;
    const int n1 = 16 + mN;
    const bool n1v = (n1 < OUT);
    const int mbase = row0 + ((lane >> 4) ? 8 : 0);
    const float bias0 = bias[n0];
    const float bias1 = n1v ? bias[n1] : 0.0f;
#pragma unroll
    for (int r = 0; r < 8; ++r) {
        const int M = mbase + r;
        out[(size_t)M * OUT + n0] = c0[r] + bias0;          // lanes 0-15 coalesced
        if (n1v) out[(size_t)M * OUT + n1] = c1[r] + bias1; // cols 16-19
    }
}

// ---------------------------------------------------------------------------
// Launcher
// Inputs (setup_inputs order): x[B*L] int, lengths[B] int,
//   emb_table[VOCAB*DIM] f32, W[OUT*DIM] f32, b[OUT] f32. Output: [B*OUT] f32.
// Workspace: rep[B][KPAD] f32 (1 MB for B=4096) + Wpad[NPAD][KPAD] f32 (8 KB).
// ---------------------------------------------------------------------------
extern "C" void kernel_launch(void* const* d_in, const int* in_sizes, int n_in,
                              void* d_out, int out_size, void* d_ws, size_t ws_size,
                              hipStream_t stream) {
    const int*   x       = (const int*)d_in[0];
    const int*   lengths = (const int*)d_in[1];
    const float* emb     = (const float*)d_in[2];
    const float* W       = (const float*)d_in[3];
    const float* bias    = (const float*)d_in[4];
    float*       out     = (float*)d_out;

    const int B = in_sizes[1];              // 4096
    const int L = in_sizes[0] / B;          // 512

    float* rep  = (float*)d_ws;                       // [B][KPAD]
    float* Wpad = rep + (size_t)B * KPAD;             // [NPAD][KPAD]

    pad_w_kernel<<<1, 256, 0, stream>>>(W, Wpad);

    dim3 blk1(64, 4);
    pool_kernel<<<B, blk1, 0, stream>>>(x, lengths, emb, rep, L);

    head_gemm_kernel<<<B / 16, 32, 0, stream>>>(rep, Wpad, bias, out);
}